// B_NNs_34789235097695
// MI455X (gfx1250) — compile-verified
//
#include <hip/hip_runtime.h>

typedef __attribute__((ext_vector_type(2))) float v2f;
typedef __attribute__((ext_vector_type(8))) float v8f;

#define HDIM 128
#define HS   132                      // padded LDS row stride (floats) -> conflict-free b64 frag loads
#define ROWS_PER_WAVE 16
#define WAVES_PER_WG  8
#define ROWS_PER_WG   (ROWS_PER_WAVE*WAVES_PER_WG)   // 128

#define BASE_RADIUS 0.06f
#define EE_RADIUS   0.045f
#define LOWER_ARM   0.176f

// Branchless tanh: 1 - 2/(exp(2x)+1).  v_exp_f32 + v_rcp_f32, no divergence.
// Saturates correctly: x>>0 -> exp=inf -> rcp=0 -> 1;  x<<0 -> exp=0 -> 1-2 = -1.
__device__ __forceinline__ float fast_tanh(float x) {
    float e = __expf(2.0f * x);
    float r = __builtin_amdgcn_rcpf(e + 1.0f);
    return fmaf(-2.0f, r, 1.0f);
}

__global__ __launch_bounds__(256, 1) void delta_mlp_kernel(
    const float* __restrict__ q,  const float* __restrict__ s,
    const float* __restrict__ sD,
    const float* __restrict__ W0, const float* __restrict__ b0,
    const float* __restrict__ W1, const float* __restrict__ b1,
    const float* __restrict__ W2, const float* __restrict__ b2,
    const float* __restrict__ W3, const float* __restrict__ b3,
    float* __restrict__ out)
{
    __shared__ float Ws[HDIM * HS];                           // W transposed: Ws[n][k], reused W1->W2
    __shared__ float hbuf[WAVES_PER_WG * ROWS_PER_WAVE * HS]; // per-wave activations

    const int t    = threadIdx.x;
    const int lane = t & 31;
    const int wave = t >> 5;
    const int l16  = lane & 15;
    const int hi   = lane >> 4;        // 0: lanes 0-15, 1: lanes 16-31
    const int kk   = hi * 2;           // K sub-offset inside a 16x16x4 fragment
    const int row0 = blockIdx.x * ROWS_PER_WG + wave * ROWS_PER_WAVE;

    float* hw = &hbuf[wave * ROWS_PER_WAVE * HS];

    // ---- cooperative load W1 -> Ws (transposed [n][k]) ----
    #pragma unroll
    for (int i = 0; i < (HDIM*HDIM)/256; ++i) {
        int idx = i*256 + t;
        int k = idx >> 7, n = idx & (HDIM-1);
        Ws[n*HS + k] = W1[idx];
    }

    // ---- layer 0: h = tanh(sD @ W0 + b0), K=3 (VALU) ----
    {
        float w0v[4], w1v[4], w2v[4], bbv[4];
        #pragma unroll
        for (int nb = 0; nb < 4; ++nb) {
            int n = nb*32 + lane;
            w0v[nb] = W0[n]; w1v[nb] = W0[HDIM + n]; w2v[nb] = W0[2*HDIM + n];
            bbv[nb] = b0[n];
        }
        for (int r = 0; r < ROWS_PER_WAVE; ++r) {
            const float* x = &sD[(size_t)(row0 + r) * 3];
            float x0 = x[0], x1 = x[1], x2 = x[2];
            #pragma unroll
            for (int nb = 0; nb < 4; ++nb) {
                int n = nb*32 + lane;
                hw[r*HS + n] = fast_tanh(fmaf(x0, w0v[nb],
                                        fmaf(x1, w1v[nb],
                                        fmaf(x2, w2v[nb], bbv[nb]))));
            }
        }
    }
    __syncthreads();   // Ws(W1) ready for all waves

    const int aBase = l16*HS + kk;     // A-frag base: row l16, K offset kk -> ds_load_b64

    // ---- hidden layer: h = tanh(h @ W + b), fp32 WMMA 16x16x4, in-place on hw ----
    auto hiddenLayer = [&](const float* __restrict__ bias) {
        v8f acc[8];
        #pragma unroll
        for (int nt = 0; nt < 8; ++nt) {           // C init = bias (column fixed per lane)
            float b = bias[nt*16 + l16];
            acc[nt] = v8f{b,b,b,b,b,b,b,b};
        }
        for (int k0 = 0; k0 < HDIM; k0 += 4) {
            v2f a = *(const v2f*)&hw[aBase + k0];               // conflict-free b64
            #pragma unroll
            for (int nt = 0; nt < 8; ++nt) {
                v2f b = *(const v2f*)&Ws[(nt*16 + l16)*HS + kk + k0];  // conflict-free b64
                acc[nt] = __builtin_amdgcn_wmma_f32_16x16x4_f32(
                    false, a, false, b, (short)0, acc[nt], false, false);
            }
        }
        // all reads of hw complete -> safe in-place writeback with tanh
        #pragma unroll
        for (int nt = 0; nt < 8; ++nt) {
            #pragma unroll
            for (int v = 0; v < 8; ++v) {
                int r = v + hi*8;                  // D layout: VGPR v -> M = v (+8 for hi lanes)
                hw[r*HS + nt*16 + l16] = fast_tanh(acc[nt][v]);
            }
        }
    };

    hiddenLayer(b1);
    __syncthreads();   // everyone done reading Ws(W1)
    #pragma unroll
    for (int i = 0; i < (HDIM*HDIM)/256; ++i) {
        int idx = i*256 + t;
        int k = idx >> 7, n = idx & (HDIM-1);
        Ws[n*HS + k] = W2[idx];
    }
    __syncthreads();   // Ws(W2) ready
    hiddenLayer(b2);

    // ---- layer 3 (N=3): split-k dot over all 32 lanes, k = 2j + hi ----
    // LDS bank for hw read: (4*l16 + 2j + hi) mod 64 -> all 32 lanes distinct.
    float o0 = hi ? 0.0f : b3[0];
    float o1 = hi ? 0.0f : b3[1];
    float o2 = hi ? 0.0f : b3[2];
    {
        const int r = l16;
        for (int j = 0; j < HDIM/2; ++j) {
            int k = 2*j + hi;
            float hv = hw[r*HS + k];
            o0 = fmaf(hv, W3[3*k+0], o0);
            o1 = fmaf(hv, W3[3*k+1], o1);
            o2 = fmaf(hv, W3[3*k+2], o2);
        }
        // fold lane-halves: lanes 0-15 end with the full sums
        o0 += __shfl_xor(o0, 16, 32);
        o1 += __shfl_xor(o1, 16, 32);
        o2 += __shfl_xor(o2, 16, 32);
    }

    // ---- Kdiag + 3x3 Cramer solve: lanes 0..15, one row each ----
    if (lane < 16) {
        const size_t row = (size_t)row0 + l16;
        float s0 = s[row*3+0], s1 = s[row*3+1], s2 = s[row*3+2];
        float q0 = q[row*3+0], q1 = q[row*3+1], q2 = q[row*3+2];

        // alpha = deg2rad([-30, 90, 210]) in fp32; compiler constant-folds
        const float al0 = -0.5235987755982988f, al1 = 1.5707963267948966f, al2 = 3.665191429188092f;
        const float ca0 = cosf(al0), ca1 = cosf(al1), ca2 = cosf(al2);
        const float sa0 = sinf(al0), sa1 = sinf(al1), sa2 = sinf(al2);
        float cq0 = __cosf(q0), cq1 = __cosf(q1), cq2 = __cosf(q2);
        float sq0 = __sinf(q0), sq1 = __sinf(q1), sq2 = __sinf(q2);

        const float dRR = BASE_RADIUS - EE_RADIUS;   // +0.015
        float Kd0 = fmaf(s0, ca0, fmaf(s1, sa0, dRR)) * sq0 - s2*cq0;
        float Kd1 = fmaf(s0, ca1, fmaf(s1, sa1, dRR)) * sq1 - s2*cq1;
        float Kd2 = fmaf(s0, ca2, fmaf(s1, sa2, dRR)) * sq2 - s2*cq2;

        const float dR = EE_RADIUS - BASE_RADIUS;    // -0.015
        // A[i][j]: row i = component, col j = arm
        float a00 = s0 + dR*ca0 - LOWER_ARM*ca0*cq0;
        float a01 = s0 + dR*ca1 - LOWER_ARM*ca1*cq1;
        float a02 = s0 + dR*ca2 - LOWER_ARM*ca2*cq2;
        float a10 = s1 + dR*sa0 - LOWER_ARM*sa0*cq0;
        float a11 = s1 + dR*sa1 - LOWER_ARM*sa1*cq1;
        float a12 = s1 + dR*sa2 - LOWER_ARM*sa2*cq2;
        float a20 = s2 - LOWER_ARM*cq0;
        float a21 = s2 - LOWER_ARM*cq1;
        float a22 = s2 - LOWER_ARM*cq2;

        float m00 = a11*a22 - a12*a21;
        float m01 = a10*a22 - a12*a20;
        float m02 = a10*a21 - a11*a20;
        float det = a00*m00 - a01*m01 + a02*m02;
        float inv = 1.0f / det;

        float x0 = (o0*m00 - a01*(o1*a22 - a12*o2) + a02*(o1*a21 - a11*o2)) * inv;
        float x1 = (a00*(o1*a22 - a12*o2) - o0*m01 + a02*(a10*o2 - o1*a20)) * inv;
        float x2 = (a00*(a11*o2 - o1*a21) - a01*(a10*o2 - o1*a20) + o0*m02) * inv;

        out[row*3+0] = Kd0 * x0;
        out[row*3+1] = Kd1 * x1;
        out[row*3+2] = Kd2 * x2;
    }
}

extern "C" void kernel_launch(void* const* d_in, const int* in_sizes, int n_in,
                              void* d_out, int out_size, void* d_ws, size_t ws_size,
                              hipStream_t stream) {
    const float* q  = (const float*)d_in[0];
    const float* s  = (const float*)d_in[1];
    const float* sD = (const float*)d_in[2];
    const float* W0 = (const float*)d_in[3];
    const float* b0 = (const float*)d_in[4];
    const float* W1 = (const float*)d_in[5];
    const float* b1 = (const float*)d_in[6];
    const float* W2 = (const float*)d_in[7];
    const float* b2 = (const float*)d_in[8];
    const float* W3 = (const float*)d_in[9];
    const float* b3 = (const float*)d_in[10];
    int B = in_sizes[0] / 3;
    int grid = B / ROWS_PER_WG;              // 262144/128 = 2048
    delta_mlp_kernel<<<grid, 256, 0, stream>>>(q, s, sD, W0, b0, W1, b1, W2, b2, W3, b3,
                                               (float*)d_out);
}